// QuantumPureQCNN_8091718386474
// MI455X (gfx1250) — compile-verified
//
#include <hip/hip_runtime.h>

#define QDIM 65536
#define QBATCH 32

typedef float v2f __attribute__((ext_vector_type(2)));
typedef float v8f __attribute__((ext_vector_type(8)));

// ---------------- complex helpers ----------------
struct cpx { float x, y; };
__device__ __forceinline__ cpx cmk(float a, float b) { cpx r; r.x = a; r.y = b; return r; }
__device__ __forceinline__ cpx cmul(cpx a, cpx b) { return cmk(a.x*b.x - a.y*b.y, a.x*b.y + a.y*b.x); }
__device__ __forceinline__ cpx cadd(cpx a, cpx b) { return cmk(a.x + b.x, a.y + b.y); }
__device__ __forceinline__ cpx csc (float s, cpx a) { return cmk(s*a.x, s*a.y); }

__device__ void mm2(const cpx A[2][2], const cpx B[2][2], cpx C[2][2]) {
    for (int i = 0; i < 2; ++i)
        for (int j = 0; j < 2; ++j)
            C[i][j] = cadd(cmul(A[i][0], B[0][j]), cmul(A[i][1], B[1][j]));
}
__device__ void mm4(const cpx A[4][4], const cpx B[4][4], cpx C[4][4]) {
    for (int i = 0; i < 4; ++i)
        for (int j = 0; j < 4; ++j) {
            cpx acc = cmk(0.f, 0.f);
            for (int k = 0; k < 4; ++k) acc = cadd(acc, cmul(A[i][k], B[k][j]));
            C[i][j] = acc;
        }
}

// R = RZ(t)*RY(t)*RX(t)  (RX applied first)
__device__ void buildR(float t, cpx R[2][2]) {
    float c = cosf(0.5f*t), s = sinf(0.5f*t);
    cpx RX[2][2]; cpx RY[2][2];
    RX[0][0] = cmk(c,0.f); RX[0][1] = cmk(0.f,-s); RX[1][0] = cmk(0.f,-s); RX[1][1] = cmk(c,0.f);
    RY[0][0] = cmk(c,0.f); RY[0][1] = cmk(-s,0.f); RY[1][0] = cmk(s,0.f);  RY[1][1] = cmk(c,0.f);
    cpx M[2][2]; mm2(RY, RX, M);
    cpx e0 = cmk(c,-s), e1 = cmk(c,s);               // diag of RZ
    R[0][0] = cmul(e0,M[0][0]); R[0][1] = cmul(e0,M[0][1]);
    R[1][0] = cmul(e1,M[1][0]); R[1][1] = cmul(e1,M[1][1]);
}

// G = (RY(tp) (x) I) * CNOT * ZZ(tz) * YY(ty) * XX(tx)   on a qubit pair (first qubit = MSB)
__device__ void buildG(float tx, float ty, float tz, float tp, cpx G[4][4]) {
    float cx = cosf(0.5f*tx), sx = sinf(0.5f*tx);
    float cy = cosf(0.5f*ty), sy = sinf(0.5f*ty);
    float cz = cosf(0.5f*tz), sz = sinf(0.5f*tz);
    float cp = cosf(0.5f*tp), sp = sinf(0.5f*tp);
    cpx XX[4][4] = {}; cpx YY[4][4] = {}; cpx ZZ[4][4] = {};
    for (int i = 0; i < 4; ++i) { XX[i][i] = cmk(cx,0.f); YY[i][i] = cmk(cy,0.f); }
    XX[0][3] = XX[1][2] = XX[2][1] = XX[3][0] = cmk(0.f,-sx);
    YY[1][2] = YY[2][1] = cmk(0.f,-sy);
    YY[0][3] = YY[3][0] = cmk(0.f, sy);
    ZZ[0][0] = ZZ[3][3] = cmk(cz,-sz);
    ZZ[1][1] = ZZ[2][2] = cmk(cz, sz);
    cpx T1[4][4], T2[4][4], T3[4][4];
    mm4(YY, XX, T1);
    mm4(ZZ, T1, T2);
    const int sig[4] = {0,1,3,2};                     // CNOT row permutation
    for (int i = 0; i < 4; ++i) for (int j = 0; j < 4; ++j) T3[i][j] = T2[sig[i]][j];
    for (int j = 0; j < 4; ++j) {                     // (RY (x) I) on the left
        G[0][j] = cadd(csc(cp,T3[0][j]), csc(-sp,T3[2][j]));
        G[1][j] = cadd(csc(cp,T3[1][j]), csc(-sp,T3[3][j]));
        G[2][j] = cadd(csc(sp,T3[0][j]), csc( cp,T3[2][j]));
        G[3][j] = cadd(csc(sp,T3[1][j]), csc( cp,T3[3][j]));
    }
}

// One thread builds the fused 16x16 complex matrix for (batch b, group g):
//   M = [Block2 pair (4g,4g+2)] * [Block1 pairs (4g,4g+1),(4g+2,4g+3)] * [R_{4g} (x) ... (x) R_{4g+3}]
__global__ __launch_bounds__(32) void build_mats(const float* __restrict__ x,
                                                 const float* __restrict__ w,
                                                 float* __restrict__ mats) {
    int idx = blockIdx.x * 32 + threadIdx.x;   // 0..127
    int b = idx >> 2, g = idx & 3;

    cpx R[4][2][2];
    for (int q = 0; q < 4; ++q) buildR(x[b*16 + 4*g + q], R[q]);

    cpx Mrot[16][16];
    for (int i = 0; i < 16; ++i)
        for (int j = 0; j < 16; ++j)
            Mrot[i][j] = cmul(cmul(R[0][(i>>3)&1][(j>>3)&1], R[1][(i>>2)&1][(j>>2)&1]),
                              cmul(R[2][(i>>1)&1][(j>>1)&1], R[3][ i     &1][ j     &1]));

    cpx GA[4][4], GB[4][4], G2[4][4];
    buildG(w[6*g+0], w[6*g+1], w[6*g+2], w[24+2*g+0], GA);   // block1 pair (4g,4g+1)
    buildG(w[6*g+3], w[6*g+4], w[6*g+5], w[24+2*g+1], GB);   // block1 pair (4g+2,4g+3)
    buildG(w[32+3*g], w[32+3*g+1], w[32+3*g+2], w[44+g], G2);// block2 pair (4g,4g+2)

    // T = Mb1 * Mrot,  Mb1[i][k] = GA[i>>2][k>>2] * GB[i&3][k&3]
    cpx T[16][16];
    for (int i = 0; i < 16; ++i)
        for (int j = 0; j < 16; ++j) {
            cpx acc = cmk(0.f,0.f);
            for (int k = 0; k < 16; ++k)
                acc = cadd(acc, cmul(cmul(GA[i>>2][k>>2], GB[i&3][k&3]), Mrot[k][j]));
            T[i][j] = acc;
        }

    // M = Mb2 * T, Mb2 acts on nibble bits (3,1), identity on (2,0)
    float* Mr = mats + (size_t)idx * 512;
    float* Mi = Mr + 256;
    for (int i = 0; i < 16; ++i) {
        int ri = ((i>>2)&2) | ((i>>1)&1);
        for (int j = 0; j < 16; ++j) {
            cpx acc = cmk(0.f,0.f);
            for (int a = 0; a < 4; ++a) {
                int k = (((a>>1)&1)<<3) | (i&4) | ((a&1)<<1) | (i&1);
                acc = cadd(acc, cmul(G2[ri][a], T[k][j]));
            }
            Mr[i*16+j] = acc.x;
            Mi[i*16+j] = acc.y;
        }
    }
}

__device__ __forceinline__ v8f wmma4(v2f a, v2f b, v8f c) {
    return __builtin_amdgcn_wmma_f32_16x16x4_f32(false, a, false, b, (short)0, c, false, false);
}

// amp of (H^16 + CZ-chain)|0>:  2^-8 * (-1)^popcount(i & (i>>1))  (real)
__device__ __forceinline__ float init_amp(int i) {
    return (__popc(i & (i >> 1)) & 1) ? -0.00390625f : 0.00390625f;
}

// ---------------------------------------------------------------------------
// Stage A: init + M0 (bits 15..12) + M1 (bits 11..8), fused through LDS.
// Block = (batch b, lo-block L): chunk of 256(hi) x 16(lo = L*16+n) amplitudes.
// Only memory traffic: one coalesced write of the chunk.
// LDS t1 layout: [p][q][n] with per-p stride 260 (bank-conflict-free both phases).
// ---------------------------------------------------------------------------
__global__ __launch_bounds__(256) void stageA(float* __restrict__ outp,
                                              const float* __restrict__ mats) {
    int blk  = blockIdx.x;            // 0..511
    int b    = blk >> 4;
    int L    = blk & 15;
    int wv   = threadIdx.x >> 5;      // wave 0..7
    int lane = threadIdx.x & 31;
    int n = lane & 15, h = lane >> 4;

    __shared__ float t1re[16 * 260];
    __shared__ float t1im[16 * 260];

    const float* Mr0 = mats + (size_t)(b*4 + 0) * 512; const float* Mi0 = Mr0 + 256;
    const float* Mr1 = mats + (size_t)(b*4 + 1) * 512; const float* Mi1 = Mr1 + 256;

    // ---- phase 1: M0, contract p' = bits 15..12; sub-tile per q = bits 11..8
    v2f ar0[4], ai0[4];
    #pragma unroll
    for (int s = 0; s < 4; ++s) {
        int k0 = 4*s + 2*h;
        ar0[s][0] = Mr0[n*16 + k0]; ar0[s][1] = Mr0[n*16 + k0 + 1];
        ai0[s][0] = Mi0[n*16 + k0]; ai0[s][1] = Mi0[n*16 + k0 + 1];
    }
    #pragma unroll
    for (int qq = 0; qq < 2; ++qq) {
        int q = 2*wv + qq;
        v2f bre[4];
        #pragma unroll
        for (int s = 0; s < 4; ++s) {
            int k0 = 4*s + 2*h;
            int i0 = (k0 << 12) | (q << 8) | (L << 4) | n;
            bre[s][0] = init_amp(i0);
            bre[s][1] = init_amp(i0 + 4096);
        }
        v8f dre = {0.f,0.f,0.f,0.f,0.f,0.f,0.f,0.f};
        v8f dim = {0.f,0.f,0.f,0.f,0.f,0.f,0.f,0.f};
        #pragma unroll
        for (int s = 0; s < 4; ++s) dre = wmma4(ar0[s], bre[s], dre);
        #pragma unroll
        for (int s = 0; s < 4; ++s) dim = wmma4(ai0[s], bre[s], dim);
        #pragma unroll
        for (int r = 0; r < 8; ++r) {
            int p = r + 8*h;
            t1re[p*260 + q*16 + n] = dre[r];
            t1im[p*260 + q*16 + n] = dim[r];
        }
    }
    __syncthreads();

    // ---- phase 2: M1, contract q' = bits 11..8; sub-tile per p = bits 15..12
    v2f ar1[4], ai1[4], an1[4];
    #pragma unroll
    for (int s = 0; s < 4; ++s) {
        int k0 = 4*s + 2*h;
        ar1[s][0] = Mr1[n*16 + k0]; ar1[s][1] = Mr1[n*16 + k0 + 1];
        ai1[s][0] = Mi1[n*16 + k0]; ai1[s][1] = Mi1[n*16 + k0 + 1];
        an1[s] = -ai1[s];
    }
    float* ore = outp + (size_t)b * QDIM;
    float* oim = outp + (size_t)(QBATCH + b) * QDIM;
    #pragma unroll
    for (int pp = 0; pp < 2; ++pp) {
        int p = 2*wv + pp;
        v2f bre[4], bim[4];
        #pragma unroll
        for (int s = 0; s < 4; ++s) {
            int k0 = 4*s + 2*h;
            bre[s][0] = t1re[p*260 + k0*16 + n]; bre[s][1] = t1re[p*260 + (k0+1)*16 + n];
            bim[s][0] = t1im[p*260 + k0*16 + n]; bim[s][1] = t1im[p*260 + (k0+1)*16 + n];
        }
        v8f dre = {0.f,0.f,0.f,0.f,0.f,0.f,0.f,0.f};
        v8f dim = {0.f,0.f,0.f,0.f,0.f,0.f,0.f,0.f};
        #pragma unroll
        for (int s = 0; s < 4; ++s) dre = wmma4(ar1[s], bre[s], dre);
        #pragma unroll
        for (int s = 0; s < 4; ++s) dre = wmma4(an1[s], bim[s], dre);
        #pragma unroll
        for (int s = 0; s < 4; ++s) dim = wmma4(ai1[s], bre[s], dim);
        #pragma unroll
        for (int s = 0; s < 4; ++s) dim = wmma4(ar1[s], bim[s], dim);
        #pragma unroll
        for (int r = 0; r < 8; ++r) {
            int q = r + 8*h;
            int a = (p << 12) | (q << 8) | (L << 4) | n;
            ore[a] = dre[r];
            oim[a] = dim[r];
        }
    }
}

// ---------------------------------------------------------------------------
// Stage B: M2 (bits 7..4) + M3 (bits 3..0) + Z-readout, fused.
// One wave owns one (b, hi) pair = 256 contiguous amplitudes. Intermediate
// transposed through a per-wave LDS tile (stride 20 -> conflict-free).
// Sign bits: bit15=(hi>>7)&1 (q0), bit11=(hi>>3)&1 (q4), bit7=(n>>3)&1 (q8), bit3=h (q12).
// ---------------------------------------------------------------------------
__global__ __launch_bounds__(256) void stageB(const float* __restrict__ in,
                                              const float* __restrict__ mats,
                                              float* __restrict__ part) {
    int wid  = (blockIdx.x << 3) | (threadIdx.x >> 5);
    int tid  = threadIdx.x;
    int wv   = threadIdx.x >> 5;
    int lane = threadIdx.x & 31;
    int b  = wid >> 8;
    int hi = wid & 255;
    int n = lane & 15, h = lane >> 4;

    __shared__ float tre[8][16 * 20];
    __shared__ float tim[8][16 * 20];

    const float* re = in + (size_t)b * QDIM;
    const float* im = in + (size_t)(QBATCH + b) * QDIM;
    const float* Mr2 = mats + (size_t)(b*4 + 2) * 512; const float* Mi2 = Mr2 + 256;
    const float* Mr3 = mats + (size_t)(b*4 + 3) * 512; const float* Mi3 = Mr3 + 256;

    // ---- M2: contract k' = bits 7..4, n = bits 3..0; loads are contiguous per wave
    v2f ar[4], ai[4], an[4], bre[4], bim[4];
    #pragma unroll
    for (int s = 0; s < 4; ++s) {
        int k0 = 4*s + 2*h;
        ar[s][0] = Mr2[n*16 + k0]; ar[s][1] = Mr2[n*16 + k0 + 1];
        ai[s][0] = Mi2[n*16 + k0]; ai[s][1] = Mi2[n*16 + k0 + 1];
        an[s] = -ai[s];
        int a0 = (hi << 8) + k0*16 + n;
        bre[s][0] = re[a0]; bre[s][1] = re[a0 + 16];
        bim[s][0] = im[a0]; bim[s][1] = im[a0 + 16];
    }
    v8f dre = {0.f,0.f,0.f,0.f,0.f,0.f,0.f,0.f};
    v8f dim = {0.f,0.f,0.f,0.f,0.f,0.f,0.f,0.f};
    #pragma unroll
    for (int s = 0; s < 4; ++s) dre = wmma4(ar[s], bre[s], dre);
    #pragma unroll
    for (int s = 0; s < 4; ++s) dre = wmma4(an[s], bim[s], dre);
    #pragma unroll
    for (int s = 0; s < 4; ++s) dim = wmma4(ai[s], bre[s], dim);
    #pragma unroll
    for (int s = 0; s < 4; ++s) dim = wmma4(ar[s], bim[s], dim);

    // stage transposed: t[a = bits7..4][c = bits3..0]
    #pragma unroll
    for (int r = 0; r < 8; ++r) {
        int a = r + 8*h;
        tre[wv][a*20 + n] = dre[r];
        tim[wv][a*20 + n] = dim[r];
    }
    __syncthreads();

    // ---- M3: contract k' = bits 3..0, n' = bits 7..4; B2[k'][n'] = t[n'][k']
    #pragma unroll
    for (int s = 0; s < 4; ++s) {
        int k0 = 4*s + 2*h;
        ar[s][0] = Mr3[n*16 + k0]; ar[s][1] = Mr3[n*16 + k0 + 1];
        ai[s][0] = Mi3[n*16 + k0]; ai[s][1] = Mi3[n*16 + k0 + 1];
        an[s] = -ai[s];
        bre[s][0] = tre[wv][n*20 + k0]; bre[s][1] = tre[wv][n*20 + k0 + 1];
        bim[s][0] = tim[wv][n*20 + k0]; bim[s][1] = tim[wv][n*20 + k0 + 1];
    }
    v8f ere = {0.f,0.f,0.f,0.f,0.f,0.f,0.f,0.f};
    v8f eim = {0.f,0.f,0.f,0.f,0.f,0.f,0.f,0.f};
    #pragma unroll
    for (int s = 0; s < 4; ++s) ere = wmma4(ar[s], bre[s], ere);
    #pragma unroll
    for (int s = 0; s < 4; ++s) ere = wmma4(an[s], bim[s], ere);
    #pragma unroll
    for (int s = 0; s < 4; ++s) eim = wmma4(ai[s], bre[s], eim);
    #pragma unroll
    for (int s = 0; s < 4; ++s) eim = wmma4(ar[s], bim[s], eim);

    float psum = 0.f;
    #pragma unroll
    for (int r = 0; r < 8; ++r) psum += ere[r]*ere[r] + eim[r]*eim[r];

    float s0 = ((hi >> 7) & 1) ? -psum : psum;   // qubit 0  (bit 15)
    float s1 = ((hi >> 3) & 1) ? -psum : psum;   // qubit 4  (bit 11)
    float s2 = ((n  >> 3) & 1) ? -psum : psum;   // qubit 8  (bit 7)
    float s3 = h ? -psum : psum;                 // qubit 12 (bit 3)

    __shared__ float sm[256][4];
    sm[tid][0] = s0; sm[tid][1] = s1; sm[tid][2] = s2; sm[tid][3] = s3;
    __syncthreads();
    for (int off = 128; off > 0; off >>= 1) {
        if (tid < off)
            for (int m = 0; m < 4; ++m) sm[tid][m] += sm[tid + off][m];
        __syncthreads();
    }
    if (tid < 4) part[blockIdx.x * 4 + tid] = sm[0][tid];
}

// part[1024][4] -> out[32][9]:  block p belongs to batch p>>5.
__global__ __launch_bounds__(288) void finalize(const float* __restrict__ part,
                                                const float* __restrict__ W,
                                                const float* __restrict__ bias,
                                                float* __restrict__ out) {
    int tid = threadIdx.x;           // 0..287
    int b = tid / 9, j = tid % 9;
    float q0 = 0.f, q1 = 0.f, q2 = 0.f, q3 = 0.f;
    for (int p = 0; p < 32; ++p) {
        const float* pp = part + (size_t)(b * 32 + p) * 4;
        q0 += pp[0]; q1 += pp[1]; q2 += pp[2]; q3 += pp[3];
    }
    out[b*9 + j] = q0*W[j*4+0] + q1*W[j*4+1] + q2*W[j*4+2] + q3*W[j*4+3] + bias[j];
}

extern "C" void kernel_launch(void* const* d_in, const int* in_sizes, int n_in,
                              void* d_out, int out_size, void* d_ws, size_t ws_size,
                              hipStream_t stream) {
    const float* x    = (const float*)d_in[0];   // (32,16)
    const float* w    = (const float*)d_in[1];   // (78,)
    const float* W    = (const float*)d_in[2];   // (9,4)
    const float* bias = (const float*)d_in[3];   // (9,)
    float* out = (float*)d_out;                  // (32,9)

    float* ws   = (float*)d_ws;
    float* bufA = ws;                                  // 16 MB intermediate state
    float* mats = ws + 2u * QBATCH * QDIM;             // 128*512 floats = 256 KB
    float* part = mats + 128 * 512;                    // 1024*4 floats = 16 KB

    build_mats<<<4, 32, 0, stream>>>(x, w, mats);
    stageA<<<512, 256, 0, stream>>>(bufA, mats);            // |0> -> M0 -> M1 (write only)
    stageB<<<1024, 256, 0, stream>>>(bufA, mats, part);     // M2 -> M3 -> readout (read only)
    finalize<<<1, 288, 0, stream>>>(part, W, bias, out);
}